// AdaLabLoss_37477884625082
// MI455X (gfx1250) — compile-verified
//
#include <hip/hip_runtime.h>
#include <stdint.h>
#include <math.h>

// AdaLabLoss for MI455X (gfx1250).
// One 1024-thread workgroup per row. The 50257-float label row (201KB) is
// staged into LDS via CDNA5 async global->LDS copies (ASYNCcnt path); while
// the DMA is in flight the kernel streams the `output` row from HBM (omax,
// o[target]) so the copy latency is fully hidden. The top-500 threshold is
// then found with a 12/12/8-bit radix select entirely out of LDS, and the
// final pass touches only ~500 L2-resident output entries. HBM traffic
// ~= 824MB total -> ~35us roofline at 23.3 TB/s.

#define TOP_TAIL 500u
#define MARGIN   0.2f

__device__ __forceinline__ unsigned sortkey_u(unsigned u) {
  // monotonic float -> uint map (ascending)
  return u ^ ((u >> 31) ? 0xFFFFFFFFu : 0x80000000u);
}

__device__ __forceinline__ float wave_sum(float v) {
#pragma unroll
  for (int o = 16; o > 0; o >>= 1) v += __shfl_down(v, o, 32);
  return v;
}
__device__ __forceinline__ float wave_max(float v) {
#pragma unroll
  for (int o = 16; o > 0; o >>= 1) v = fmaxf(v, __shfl_down(v, o, 32));
  return v;
}
__device__ __forceinline__ void wave_top2(float& m1, float& m2) {
#pragma unroll
  for (int o = 16; o > 0; o >>= 1) {
    float b1 = __shfl_down(m1, o, 32);
    float b2 = __shfl_down(m2, o, 32);
    if (b1 > m1) { m2 = fmaxf(m1, b2); m1 = b1; }
    else         { m2 = fmaxf(m2, b1); }
  }
}

// Cooperative descending-rank bin select: finds bin b (scanning from the top)
// where the suffix count first reaches ctrl[1]; leaves b in ctrl[0] and the
// residual rank (rank within bin b) in ctrl[1]. All threads must call.
__device__ void selectBin(unsigned* hist, unsigned* gsum, unsigned* ctrl,
                          int nbins, int tid, int nt) {
  int groups = nbins >> 5;
  __syncthreads();                        // hist complete
  for (int g = tid; g < groups; g += nt) {
    unsigned s = 0;
    int base = g << 5;
#pragma unroll
    for (int j = 0; j < 32; ++j) s += hist[base + j];
    gsum[g] = s;
  }
  __syncthreads();
  if (tid == 0) {
    unsigned k = ctrl[1];
    unsigned cum = 0;
    int g = groups - 1;
    for (; g > 0; --g) { if (cum + gsum[g] >= k) break; cum += gsum[g]; }
    int lo = g << 5;
    int b = lo + 31;
    for (; b > lo; --b) { if (cum + hist[b] >= k) break; cum += hist[b]; }
    ctrl[0] = (unsigned)b;
    ctrl[1] = k - cum;
  }
  __syncthreads();
}

extern "C" __global__ __launch_bounds__(1024)
void adalab_row_kernel(const float* __restrict__ output,
                       const int* __restrict__ target,
                       const float* __restrict__ label,
                       float* __restrict__ row_loss,
                       int V) {
  extern __shared__ char smem[];
  const int row  = blockIdx.x;
  const int tid  = threadIdx.x;
  const int nt   = blockDim.x;
  const int lane = tid & 31;
  const int wid  = tid >> 5;
  const int nw   = nt >> 5;
  const int tgt  = target[row];

  // LDS layout: [16 .. 16+pad+4V) row floats; then hist/gsum/red/ctrl.
  const unsigned histOff = (4u * (unsigned)V + 64u) & ~15u;
  unsigned* hist = (unsigned*)(smem + histOff);                 // 4096 u32
  unsigned* gsum = (unsigned*)(smem + histOff + 16384u);        // 128 u32
  float*    red  = (float*)  (smem + histOff + 16384u + 512u);  // 128 f32
  unsigned* ctrl = (unsigned*)(smem + histOff + 16384u + 1024u);// 8 u32
  float*    resf = (float*)  (ctrl + 8);                        // 8 f32

  const float* gp = label + (size_t)row * (size_t)V;
  const unsigned pad = (unsigned)((uintptr_t)gp & 15u);         // 0,4,8,12
  char* rowBase = smem + 16;  // element i lives at rowBase + pad + 4*i

  // ---- Stage label row into LDS with CDNA5 async global->LDS copies ----
  {
    unsigned pre = ((16u - pad) & 15u) >> 2;   // leading floats to 16B align
    if (pre > (unsigned)V) pre = (unsigned)V;
    if ((unsigned)tid < pre) {
      unsigned lds = (unsigned)(uintptr_t)(rowBase + pad + 4u * (unsigned)tid);
      unsigned long long ga = (unsigned long long)(uintptr_t)(gp + tid);
      asm volatile("global_load_async_to_lds_b32 %0, %1, off"
                   :: "v"(lds), "v"(ga) : "memory");
    }
    unsigned nq = ((unsigned)V - pre) >> 2;    // aligned 16B quads
    for (unsigned q = (unsigned)tid; q < nq; q += (unsigned)nt) {
      unsigned e = pre + (q << 2);
      unsigned lds = (unsigned)(uintptr_t)(rowBase + pad + 4u * e);
      unsigned long long ga = (unsigned long long)(uintptr_t)(gp + e);
      asm volatile("global_load_async_to_lds_b128 %0, %1, off"
                   :: "v"(lds), "v"(ga) : "memory");
    }
    unsigned done = pre + (nq << 2);
    unsigned rem  = (unsigned)V - done;
    if ((unsigned)tid < rem) {
      unsigned e = done + (unsigned)tid;
      unsigned lds = (unsigned)(uintptr_t)(rowBase + pad + 4u * e);
      unsigned long long ga = (unsigned long long)(uintptr_t)(gp + e);
      asm volatile("global_load_async_to_lds_b32 %0, %1, off"
                   :: "v"(lds), "v"(ga) : "memory");
    }
    // NOTE: no wait yet — overlap the output-row HBM scan with the DMA.
  }

  const float NEGINF = -__builtin_inff();
  const float* orow = output + (size_t)row * (size_t)V;

  // ---- Overlapped with DMA: stream output row (omax, o[target]) ----
  {
    float omax = NEGINF;
    for (int i = tid; i < V; i += nt) {
      float o = orow[i];
      omax = fmaxf(omax, o);
      if (i == tgt) resf[1] = o;                 // o[target]
    }
    omax = wave_max(omax);
    if (lane == 0) red[wid] = omax;
    __syncthreads();
    if (tid == 0) {
      float om = NEGINF;
      for (int w = 0; w < nw; ++w) om = fmaxf(om, red[w]);
      resf[2] = om;                              // row max of output
    }
  }

  // DMA must be complete before touching the staged row.
  asm volatile("s_wait_asynccnt 0x0" ::: "memory");
  __syncthreads();

  const float* rowf = (const float*)(rowBase + pad);

  // ---- Pass 1: 12-bit histogram + fused top-2 ----
  for (int j = tid; j < 4096; j += nt) hist[j] = 0;
  if (tid == 0) ctrl[1] = TOP_TAIL;
  __syncthreads();

  float m1 = NEGINF, m2 = NEGINF;
  for (int i = tid; i < V; i += nt) {
    if (i == tgt || i == 0) continue;        // excluded entries (-inf in ref)
    float f = rowf[i];
    unsigned key = sortkey_u(__float_as_uint(f));
    atomicAdd(&hist[key >> 20], 1u);
    if (f > m1) { m2 = m1; m1 = f; } else if (f > m2) { m2 = f; }
  }
  wave_top2(m1, m2);
  if (lane == 0) { red[wid] = m1; red[32 + wid] = m2; }
  __syncthreads();
  if (wid == 0) {
    float a1 = (lane < nw) ? red[lane]      : NEGINF;
    float a2 = (lane < nw) ? red[32 + lane] : NEGINF;
    wave_top2(a1, a2);
    if (lane == 0) resf[0] = a2;             // 2nd-largest (kth_lower)
  }
  selectBin(hist, gsum, ctrl, 4096, tid, nt);
  const unsigned b1 = ctrl[0];

  // ---- Pass 2: next 12 bits within bin b1 ----
  __syncthreads();
  for (int j = tid; j < 4096; j += nt) hist[j] = 0;
  __syncthreads();
  for (int i = tid; i < V; i += nt) {
    if (i == tgt || i == 0) continue;
    unsigned key = sortkey_u(__float_as_uint(rowf[i]));
    if ((key >> 20) == b1) atomicAdd(&hist[(key >> 8) & 0xFFFu], 1u);
  }
  selectBin(hist, gsum, ctrl, 4096, tid, nt);
  const unsigned p24 = (b1 << 12) | ctrl[0];

  // ---- Pass 3: final 8 bits -> exact 500th-largest key ----
  __syncthreads();
  for (int j = tid; j < 256; j += nt) hist[j] = 0;
  __syncthreads();
  for (int i = tid; i < V; i += nt) {
    if (i == tgt || i == 0) continue;
    unsigned key = sortkey_u(__float_as_uint(rowf[i]));
    if ((key >> 8) == p24) atomicAdd(&hist[key & 255u], 1u);
  }
  selectBin(hist, gsum, ctrl, 256, tid, nt);
  const unsigned tkey = (p24 << 8) | ctrl[0];           // kth_upper key

  const float    msec = resf[0];                        // kth_lower value
  const unsigned mkey = sortkey_u(__float_as_uint(msec));

  // ---- Pass 4: softmax sums (LDS) + ~500 scattered L2-hit output reads ----
  float Z = 0.f, Bs = 0.f, Cs = 0.f;
  for (int i = tid; i < V; i += nt) {
    if (i == tgt || i == 0) continue;
    float f = rowf[i];
    unsigned key = sortkey_u(__float_as_uint(f));
    if (key >= tkey && key <= mkey) {
      float w = expf(f - msec);
      Z  += w;
      Bs += w * (f - msec);
      Cs += w * orow[i];                                // L2 hit (just streamed)
    }
  }
  Z = wave_sum(Z); Bs = wave_sum(Bs); Cs = wave_sum(Cs);
  __syncthreads();                                      // red[] reuse
  if (lane == 0) { red[wid] = Z; red[32 + wid] = Bs; red[64 + wid] = Cs; }
  __syncthreads();
  if (tid == 0) {
    float z = 0.f, bs = 0.f, cs = 0.f;
    for (int w = 0; w < nw; ++w) {
      z += red[w]; bs += red[32 + w]; cs += red[64 + w];
    }
    float loss = 0.f;
    if (tgt != 0) {
      float ot   = resf[1];
      float om   = resf[2];
      float pmax = expf(om);
      float pg   = expf(ot);
      float ub   = z / (z + 1.f) - MARGIN;              // 1/(1+vmax) - margin
      float eps  = fminf(1.f - pmax, ub);
      float r    = pg / pmax;
      eps *= r * r;                                     // alpha * epsilon
      float conf = 1.f - eps;
      float lv = (eps  > 0.f) ? eps * (logf(eps) - logf(z)) + (eps / z) * (bs - cs) : 0.f;
      float lt = (conf > 0.f) ? conf * (logf(conf) - ot) : 0.f;
      loss = lv + lt;
    }
    row_loss[row] = loss;
  }
}

extern "C" __global__ void adalab_reduce_kernel(const float* __restrict__ rl,
                                               float* __restrict__ out, int n) {
  __shared__ float s[256];
  float a = 0.f;
  for (int i = threadIdx.x; i < n; i += 256) a += rl[i];
  s[threadIdx.x] = a;
  __syncthreads();
  for (int off = 128; off > 0; off >>= 1) {
    if ((int)threadIdx.x < off) s[threadIdx.x] += s[threadIdx.x + off];
    __syncthreads();
  }
  if (threadIdx.x == 0) out[0] = s[0];
}

extern "C" void kernel_launch(void* const* d_in, const int* in_sizes, int n_in,
                              void* d_out, int out_size, void* d_ws, size_t ws_size,
                              hipStream_t stream) {
  const float* output = (const float*)d_in[0];   // [B,V] f32 log-probs
  const int*   target = (const int*)d_in[1];     // [B]   i32
  const float* label  = (const float*)d_in[2];   // [B,V] f32
  const int Bn = in_sizes[1];
  const int V  = in_sizes[0] / Bn;

  float* rl = (float*)d_ws;                      // per-row losses (B floats)

  const unsigned histOff = (4u * (unsigned)V + 64u) & ~15u;
  const size_t   shbytes = (size_t)histOff + 16384u + 1024u + 64u;

  hipFuncSetAttribute((const void*)adalab_row_kernel,
                      hipFuncAttributeMaxDynamicSharedMemorySize, (int)shbytes);

  adalab_row_kernel<<<Bn, 1024, shbytes, stream>>>(output, target, label, rl, V);
  adalab_reduce_kernel<<<1, 256, 0, stream>>>(rl, (float*)d_out, Bn);
}